// weighted_loss_7980049236598
// MI455X (gfx1250) — compile-verified
//
#include <hip/hip_runtime.h>
#include <hip/hip_bf16.h>

#define TBL_BITS 19
#define TBL (1u << TBL_BITS)            // 524288 slots > 2x N
#define EMPTY_KEY 0xFFFFFFFFFFFFFFFFULL
#define RED_BLOCKS 512
#define RED_THREADS 256
#define EPT 8                           // edges per thread in scatter kernel

__device__ __forceinline__ unsigned int mix64(unsigned long long k) {
    // MurmurHash3 fmix64 finalizer
    k ^= k >> 33; k *= 0xff51afd7ed558ccdULL;
    k ^= k >> 33; k *= 0xc4ceb9fe1a85ec53ULL;
    k ^= k >> 33;
    return (unsigned int)k;
}

// ---------------------------------------------------------------- init ws ---
__global__ void k_init(unsigned long long* __restrict__ deg,
                       unsigned long long* __restrict__ hkeys,
                       unsigned int* __restrict__ hcnt,
                       float* __restrict__ partials, int N) {
    int i = blockIdx.x * blockDim.x + threadIdx.x;
    int stride = gridDim.x * blockDim.x;
    for (int t = i; t < (int)TBL; t += stride) { hkeys[t] = EMPTY_KEY; hcnt[t] = 0u; }
    for (int t = i; t < N; t += stride) deg[t] = 0ull;
    for (int t = i; t < 2 * RED_BLOCKS; t += stride) partials[t] = 0.0f;
}

// ------------------------------------------------- edge scatter (E edges) ---
// deg[r] packs: high 32 bits = s0 (# neighbors with x==1), low 32 = s1.
// One u64 atomic per edge instead of two u32 atomics (halves L2 atomic ops).
__global__ void k_edges(const int* __restrict__ row, const int* __restrict__ col,
                        const int* __restrict__ x,
                        unsigned long long* __restrict__ deg, int E) {
    int i = blockIdx.x * blockDim.x + threadIdx.x;
    int e0 = i * EPT;
    if (e0 >= E) return;

    // gfx1250 streaming prefetch ~64KB ahead in both edge streams
    int pf = e0 + 16384;
    if (pf < E) {
        __builtin_prefetch((const void*)(row + pf), 0, 1);
        __builtin_prefetch((const void*)(col + pf), 0, 1);
    }

    if (e0 + EPT - 1 < E) {
        // two b128 loads per stream, issued before the atomic burst
        const int4* r4p = (const int4*)(row + e0);
        const int4* c4p = (const int4*)(col + e0);
        int4 ra = r4p[0], rb = r4p[1];
        int4 ca = c4p[0], cb = c4p[1];
        int rr[EPT] = { ra.x, ra.y, ra.z, ra.w, rb.x, rb.y, rb.z, rb.w };
        int cc[EPT] = { ca.x, ca.y, ca.z, ca.w, cb.x, cb.y, cb.z, cb.w };
        unsigned long long add[EPT];
#pragma unroll
        for (int j = 0; j < EPT; ++j)
            add[j] = (x[cc[j]] > 0) ? (1ull << 32) : 1ull;
#pragma unroll
        for (int j = 0; j < EPT; ++j)
            atomicAdd(&deg[rr[j]], add[j]);
    } else {
        for (int e = e0; e < E; ++e) {
            unsigned long long add = (x[col[e]] > 0) ? (1ull << 32) : 1ull;
            atomicAdd(&deg[row[e]], add);
        }
    }
}

// ------------------------------------- hash insert: count key multiplicity ---
// key = (x << 48) | (s0 << 24) | s1  (s0,s1 <= E < 2^24, x in {0,1})
__global__ void k_insert(const int* __restrict__ x,
                         const unsigned long long* __restrict__ deg,
                         unsigned long long* __restrict__ hkeys,
                         unsigned int* __restrict__ hcnt,
                         unsigned int* __restrict__ slot, int N) {
    int i = blockIdx.x * blockDim.x + threadIdx.x;
    if (i >= N) return;
    unsigned long long d = deg[i];
    unsigned long long s1 = d & 0xFFFFFFFFull;   // neighbors with x==0
    unsigned long long s0 = d >> 32;             // neighbors with x==1
    unsigned long long key = ((unsigned long long)(unsigned int)x[i] << 48)
                           | (s0 << 24) | s1;
    unsigned int h = mix64(key) & (TBL - 1u);
    for (;;) {
        unsigned long long prev = atomicCAS(&hkeys[h], EMPTY_KEY, key);
        if (prev == EMPTY_KEY || prev == key) {
            atomicAdd(&hcnt[h], 1u);
            slot[i] = h;
            break;
        }
        h = (h + 1u) & (TBL - 1u);
    }
}

// --------------------------- per-node weighted NLL + deterministic partials ---
__global__ void k_loss(const float* __restrict__ outp, const int* __restrict__ y,
                       const unsigned int* __restrict__ slot,
                       const unsigned int* __restrict__ hcnt,
                       float* __restrict__ partials, int N) {
    __shared__ float shw[RED_THREADS];
    __shared__ float shl[RED_THREADS];
    float accw = 0.0f, accl = 0.0f;
    int stride = gridDim.x * blockDim.x;
    for (int i = blockIdx.x * blockDim.x + threadIdx.x; i < N; i += stride) {
        float2 o = ((const float2*)outp)[i];
        float m = fmaxf(o.x, o.y);
        float lse = m + logf(expf(o.x - m) + expf(o.y - m));
        float oy = (y[i] > 0) ? o.y : o.x;
        float nll = lse - oy;                       // -log_softmax[y]
        float c = (float)hcnt[slot[i]];
        float w = 1.0f / sqrtf(c);                  // counts ** (-0.5)
        accw += w;
        accl += w * nll;
    }
    shw[threadIdx.x] = accw;
    shl[threadIdx.x] = accl;
    __syncthreads();
    for (int s = RED_THREADS / 2; s > 0; s >>= 1) {
        if (threadIdx.x < (unsigned)s) {
            shw[threadIdx.x] += shw[threadIdx.x + s];
            shl[threadIdx.x] += shl[threadIdx.x + s];
        }
        __syncthreads();
    }
    if (threadIdx.x == 0) {
        partials[blockIdx.x] = shw[0];
        partials[RED_BLOCKS + blockIdx.x] = shl[0];
    }
}

// ------------------------------------------------ fixed-order final reduce ---
__global__ void k_final(const float* __restrict__ partials, float* __restrict__ outv) {
    __shared__ float shw[RED_THREADS];
    __shared__ float shl[RED_THREADS];
    float aw = 0.0f, al = 0.0f;
    for (int i = threadIdx.x; i < RED_BLOCKS; i += RED_THREADS) {
        aw += partials[i];
        al += partials[RED_BLOCKS + i];
    }
    shw[threadIdx.x] = aw;
    shl[threadIdx.x] = al;
    __syncthreads();
    for (int s = RED_THREADS / 2; s > 0; s >>= 1) {
        if (threadIdx.x < (unsigned)s) {
            shw[threadIdx.x] += shw[threadIdx.x + s];
            shl[threadIdx.x] += shl[threadIdx.x + s];
        }
        __syncthreads();
    }
    if (threadIdx.x == 0) outv[0] = shl[0] / shw[0];
}

static inline size_t align16(size_t v) { return (v + 15) & ~(size_t)15; }

extern "C" void kernel_launch(void* const* d_in, const int* in_sizes, int n_in,
                              void* d_out, int out_size, void* d_ws, size_t ws_size,
                              hipStream_t stream) {
    const float* outp = (const float*)d_in[0];      // (N,2) f32
    const int*   x    = (const int*)d_in[1];        // (N,)  i32
    const int*   y    = (const int*)d_in[2];        // (N,)  i32
    const int*   ei   = (const int*)d_in[3];        // (2,E) i32 row-major

    const int N = in_sizes[1];
    const int E = in_sizes[3] / 2;
    const int* row = ei;
    const int* col = ei + E;

    // ---- workspace carve (L2-resident: ~8.7 MB total) ----
    char* p = (char*)d_ws;
    unsigned long long* deg   = (unsigned long long*)p; p += align16((size_t)N * 8);
    unsigned long long* hkeys = (unsigned long long*)p; p += align16((size_t)TBL * 8);
    unsigned int*       hcnt  = (unsigned int*)p;       p += align16((size_t)TBL * 4);
    unsigned int*       slot  = (unsigned int*)p;       p += align16((size_t)N * 4);
    float*              parts = (float*)p;              p += align16((size_t)2 * RED_BLOCKS * 4);
    (void)ws_size;

    const int T = 256;
    int initWork = (int)TBL > N ? (int)TBL : N;
    int initBlocks = (initWork + T - 1) / T;
    k_init<<<initBlocks, T, 0, stream>>>(deg, hkeys, hcnt, parts, N);

    int nq = (E + EPT - 1) / EPT;
    int edgeBlocks = (nq + T - 1) / T;
    k_edges<<<edgeBlocks, T, 0, stream>>>(row, col, x, deg, E);

    int nodeBlocks = (N + T - 1) / T;
    k_insert<<<nodeBlocks, T, 0, stream>>>(x, deg, hkeys, hcnt, slot, N);

    k_loss<<<RED_BLOCKS, RED_THREADS, 0, stream>>>(outp, y, slot, hcnt, parts, N);
    k_final<<<1, RED_THREADS, 0, stream>>>(parts, (float*)d_out);
}